// LowerTraingularSparseLayer_63041529970719
// MI455X (gfx1250) — compile-verified
//
#include <hip/hip_runtime.h>
#include <hip/hip_bf16.h>

// y = tril(W) @ x, N=8192, fp32. HBM-bound: lower triangle = 134 MB -> ~5.8us
// floor @ 23.3 TB/s. Strategy: async global->LDS streaming of 16x64 W tiles,
// V_WMMA_F32_16X16X4_F32 for the row dot-products, 2D grid over (K-chunk,
// row-block) to kill triangular tail imbalance, deterministic 2-pass reduce.

typedef __attribute__((ext_vector_type(2))) float v2f;
typedef __attribute__((ext_vector_type(8))) float v8f;

#define N        8192
#define BM       16      // rows per row-block
#define KT       512     // K super-tile staged per iteration (x tile)
#define WKT      64      // K sub-tile per wave (8 waves * 64 = 512)
#define NW       8       // waves per block (256 threads, wave32)
#define LSTRIDE  68      // LDS row stride in floats: 68 % 64 == 4 -> conflict-free
#define KCHUNKS  8       // grid.x: K split per row-block
#define KCHUNK   (N / KCHUNKS)   // 1024

__global__ __launch_bounds__(256)
void tril_matvec_wmma_partial(const float* __restrict__ W,
                              const float* __restrict__ x,
                              float* __restrict__ partial) {
    const int rb  = blockIdx.y;          // row block: rows [16*rb, 16*rb+16)
    const int r0  = rb * BM;
    const int kc0 = blockIdx.x * KCHUNK; // this block's K range
    const int kc_end = kc0 + KCHUNK;     // never exceeds N (KCHUNK divides N)

    const int tid  = threadIdx.x;
    const int wave = tid >> 5;
    const int lane = tid & 31;
    const int hf   = lane >> 4;          // half-wave select (K pair 0/1 vs 2/3)
    const int mrow = lane & 15;          // row-within-tile / N index

    __shared__ __align__(16) float xs[KT];                    // 2 KB
    __shared__ __align__(16) float wtile[NW][BM * LSTRIDE];   // 8 * 4352 B
    __shared__ float red[NW][BM];

    v8f c = {0.f, 0.f, 0.f, 0.f, 0.f, 0.f, 0.f, 0.f};

    // Uniform loop bounds: only K super-tiles that intersect the lower triangle.
    for (int k0 = kc0; k0 < kc_end && k0 <= r0 + (BM - 1); k0 += KT) {
        __syncthreads();                 // protect xs from previous readers
        xs[tid]       = x[k0 + tid];
        xs[tid + 256] = x[k0 + 256 + tid];
        __syncthreads();

        const int ks = k0 + wave * WKT;  // wave-uniform -> EXEC stays all-ones
        if (ks <= r0 + (BM - 1)) {
            // ---- async stage W[r0..r0+15, ks..ks+64) into this wave's LDS tile
            // 8 x global_load_async_to_lds_b128: each moves 2 rows x 64 floats.
            const float* gbase = W + (size_t)r0 * N + ks;
            const unsigned lbase =
                (unsigned)(unsigned long long)(void*)&wtile[wave][0];
            const int colq = (lane & 15) * 4;        // 16B-aligned column chunk
            #pragma unroll
            for (int i = 0; i < 8; ++i) {
                const int r = i * 2 + hf;
                const float* g = gbase + (size_t)r * N + colq;
                const unsigned l = lbase + (unsigned)((r * LSTRIDE + colq) * 4);
                asm volatile("global_load_async_to_lds_b128 %0, %1, off"
                             :: "v"(l), "v"(g) : "memory");
            }
            asm volatile("s_wait_asynccnt 0" ::: "memory");

            // ---- 16 WMMA steps: D[m,n] += sum_k x[k] * W[r0+n, k]
            const int xbase  = ks - k0;
            const int rowabs = r0 + mrow;
            #pragma unroll
            for (int kk = 0; kk < WKT / 4; ++kk) {
                const int cloc = kk * 4 + 2 * hf;    // K pair within sub-tile
                const int cabs = ks + cloc;
                // A fragment: x broadcast across M (lane halves pick K pairs)
                const float2 xa = *(const float2*)&xs[xbase + cloc];
                v2f a; a.x = xa.x; a.y = xa.y;
                // B fragment: W[r0+mrow, cabs..cabs+1], masked to lower triangle
                const float2 wv =
                    *(const float2*)&wtile[wave][mrow * LSTRIDE + cloc];
                v2f b;
                b.x = (cabs     <= rowabs) ? wv.x : 0.0f;
                b.y = (cabs + 1 <= rowabs) ? wv.y : 0.0f;
                c = __builtin_amdgcn_wmma_f32_16x16x4_f32(
                        false, a, false, b, (short)0, c, false, false);
            }
        }
    }

    // c[0] on lanes 0..15 = partial y[r0 + lane] for this wave's K columns.
    __syncthreads();
    if (hf == 0) red[wave][mrow] = c[0];
    __syncthreads();
    if (tid < BM) {
        float s = 0.f;
        #pragma unroll
        for (int w = 0; w < NW; ++w) s += red[w][tid];
        partial[(size_t)blockIdx.x * N + r0 + tid] = s;  // always written (zeros
    }                                                    // for inactive chunks)
}

__global__ __launch_bounds__(256)
void tril_matvec_reduce(const float* __restrict__ partial,
                        float* __restrict__ y) {
    const int i = blockIdx.x * blockDim.x + threadIdx.x;
    if (i < N) {
        float s = 0.f;
        #pragma unroll
        for (int k = 0; k < KCHUNKS; ++k) s += partial[(size_t)k * N + i];
        y[i] = s;
    }
}

extern "C" void kernel_launch(void* const* d_in, const int* in_sizes, int n_in,
                              void* d_out, int out_size, void* d_ws, size_t ws_size,
                              hipStream_t stream) {
    (void)in_sizes; (void)n_in; (void)out_size; (void)ws_size;
    const float* x = (const float*)d_in[0];   // setup_inputs order: x, then W
    const float* W = (const float*)d_in[1];
    float*       y = (float*)d_out;
    float* partial = (float*)d_ws;            // KCHUNKS * N * 4 = 256 KB scratch

    dim3 grid(KCHUNKS, N / BM);               // 8 x 512 blocks
    tril_matvec_wmma_partial<<<grid, 256, 0, stream>>>(W, x, partial);
    tril_matvec_reduce<<<dim3(N / 256), 256, 0, stream>>>(partial, y);
}